// RCNN_2224793059523
// MI455X (gfx1250) — compile-verified
//
#include <hip/hip_runtime.h>
#include <float.h>

// Problem constants (from reference setup_inputs)
#define PP    7
#define RROI  128
#define CCH   512
#define FH    37
#define FW    37
#define KFC1  (CCH * PP * PP)   // 25088
#define NHID  4096

typedef __attribute__((ext_vector_type(2))) float v2f;
typedef __attribute__((ext_vector_type(8))) float v8f;

// ---------------------------------------------------------------------------
// Kernel 1: classic Fast-RCNN RoI max pooling.
// grid = (R, C); block = 64 threads; thread t<49 computes output cell (ph,pw).
// ---------------------------------------------------------------------------
__global__ __launch_bounds__(64) void roi_pool_kernel(
    const float* __restrict__ feat, const float* __restrict__ rois,
    const int* __restrict__ batch_id, float* __restrict__ X)
{
  const int r = blockIdx.x;
  const int c = blockIdx.y;
  const int t = threadIdx.x;
  if (t >= PP * PP) return;
  const int ph = t / PP, pw = t % PP;

  const float scale = 1.0f / 16.0f;
  // jnp.round == round-half-to-even == rintf
  const int x0 = (int)rintf(rois[r * 4 + 0] * scale);
  const int y0 = (int)rintf(rois[r * 4 + 1] * scale);
  const int x1 = (int)rintf(rois[r * 4 + 2] * scale);
  const int y1 = (int)rintf(rois[r * 4 + 3] * scale);
  const float rh = (float)max(y1 - y0 + 1, 1);
  const float rw = (float)max(x1 - x0 + 1, 1);

  const int hs = min(max(y0 + (int)floorf((float)ph * rh / (float)PP), 0), FH);
  const int he = min(max(y0 + (int)ceilf((float)(ph + 1) * rh / (float)PP), 0), FH);
  const int ws = min(max(x0 + (int)floorf((float)pw * rw / (float)PP), 0), FW);
  const int we = min(max(x0 + (int)ceilf((float)(pw + 1) * rw / (float)PP), 0), FW);

  const float* src = feat + (((size_t)batch_id[r] * CCH + c) * FH) * FW;
  float m = -FLT_MAX;
  for (int h = hs; h < he; ++h)
    for (int w = ws; w < we; ++w)
      m = fmaxf(m, src[h * FW + w]);

  const bool valid = (he > hs) && (we > ws);
  X[(size_t)r * KFC1 + (size_t)c * (PP * PP) + ph * PP + pw] = valid ? m : 0.0f;
}

// ---------------------------------------------------------------------------
// Kernel 2: fp32 WMMA GEMM, fixed M=128, split-K:
//   grid = (N/64, S); slice s = blockIdx.y handles K-range [s*KS, (s+1)*KS).
//   FINAL=true  (S==1): Y[m,n] = act(acc + bias)        (heads)
//   FINAL=false       : partials Y[s][m,n] = acc        (FC1/FC2; reduced later)
// One wave = 16 columns x all 128 rows (8 x v8f accumulators); per K=4 step:
// 2 NT B loads (weights read exactly once from HBM) + 8 A loads (L2-hot)
// + 8 V_WMMA_F32_16X16X4_F32. Split-K exists purely for occupancy: 2048 waves
// for FC1 instead of 256, enough outstanding loads to saturate 23.3 TB/s.
// Weight traffic is unchanged (slices partition K).
//
// Fragment layouts per CDNA5 ISA 7.12.2 (wave32):
//   A 16x4 : lane m=lane&15; VGPR0 = K(khalf), VGPR1 = K(khalf+1), khalf=2*(lane>>4)
//   B 4x16 : lane n=lane&15; VGPR0 = row khalf, VGPR1 = row khalf+1
//   C/D    : VGPR j of lane L -> element [M = j + 8*(L>=16)][N = L&15]
// ---------------------------------------------------------------------------
template <int K, int KS, int N, bool FINAL, bool RELU>
__global__ __launch_bounds__(128) void wmma_gemm128(
    const float* __restrict__ X, const float* __restrict__ Wt,
    const float* __restrict__ bias, float* __restrict__ Y)
{
  constexpr bool EDGE = (N % 16) != 0;

  const int lane = threadIdx.x & 31;
  const int wave = threadIdx.x >> 5;
  const int n0   = blockIdx.x * 64 + wave * 16;
  if (n0 >= N) return;                  // whole-wave exit: EXEC stays all-ones
  const int s  = blockIdx.y;            // K-slice
  const int k0 = s * KS;

  const int  hi    = lane >> 4;         // 0 | 1
  const int  mrow  = lane & 15;
  const int  khalf = hi * 2;            // 0 | 2
  const int  ncol  = n0 + (lane & 15);
  const bool nv    = EDGE ? (ncol < N) : true;
  const int  ncl   = EDGE ? (nv ? ncol : (N - 1)) : ncol;  // clamped column

  v8f acc[8] = {};

  const int aoff = mrow * K + k0 + khalf;      // 32-bit lane offset into X
  const int boff = (k0 + khalf) * N + ncl;     // 32-bit lane offset into W

#pragma unroll 4
  for (int k = 0; k < KS; k += 4) {
    // B fragment: weight stream, read-once -> non-temporal (keep L2 for acts)
    float b0 = __builtin_nontemporal_load(Wt + boff + k * N);
    float b1 = __builtin_nontemporal_load(Wt + boff + k * N + N);
    v2f b;
    if (EDGE) { b.x = nv ? b0 : 0.0f; b.y = nv ? b1 : 0.0f; }
    else      { b.x = b0;             b.y = b1; }
#pragma unroll
    for (int r = 0; r < 8; ++r) {
      // A fragment, row tile r: compile-time r*16*K folds into scalar bases
      v2f a = *(const v2f*)(X + aoff + r * 16 * K + k);   // 8B aligned (K even)
      acc[r] = __builtin_amdgcn_wmma_f32_16x16x4_f32(
          /*neg_a=*/false, a, /*neg_b=*/false, b,
          /*c_mod=*/(short)0, acc[r], /*reuse_a=*/false, /*reuse_b=*/false);
    }
  }

  if (FINAL) {
    const float bv = nv ? bias[ncl] : 0.0f;
#pragma unroll
    for (int r = 0; r < 8; ++r)
#pragma unroll
      for (int j = 0; j < 8; ++j) {
        float v = acc[r][j] + bv;
        if (RELU) v = fmaxf(v, 0.0f);
        const int m = r * 16 + hi * 8 + j;
        if (!EDGE || nv) Y[m * N + ncol] = v;
      }
  } else {
    float* Yp = Y + s * (RROI * N);            // fp32 partials, no bias/act
#pragma unroll
    for (int r = 0; r < 8; ++r)
#pragma unroll
      for (int j = 0; j < 8; ++j) {
        const int m = r * 16 + hi * 8 + j;
        Yp[m * N + ncol] = acc[r][j];          // N%16==0 on the partial path
      }
  }
}

// ---------------------------------------------------------------------------
// Kernel 3: deterministic split-K reduction + bias + ReLU, float4-vectorized.
//   Yp: [S][128][N] partials -> Y: [128][N]
// ---------------------------------------------------------------------------
template <int N, int S>
__global__ __launch_bounds__(256) void reduce_bias_relu(
    const float* __restrict__ Yp, const float* __restrict__ bias,
    float* __restrict__ Y)
{
  constexpr int TOT = RROI * N / 4;            // float4 elements
  const int i = blockIdx.x * 256 + threadIdx.x;
  if (i >= TOT) return;
  float4 sum = ((const float4*)Yp)[i];
#pragma unroll
  for (int s = 1; s < S; ++s) {
    float4 p = ((const float4*)Yp)[i + s * TOT];
    sum.x += p.x; sum.y += p.y; sum.z += p.z; sum.w += p.w;
  }
  const float4 bv = ((const float4*)bias)[i % (N / 4)];
  float4 o;
  o.x = fmaxf(sum.x + bv.x, 0.0f);
  o.y = fmaxf(sum.y + bv.y, 0.0f);
  o.z = fmaxf(sum.z + bv.z, 0.0f);
  o.w = fmaxf(sum.w + bv.w, 0.0f);
  ((float4*)Y)[i] = o;
}

// ---------------------------------------------------------------------------
// Host-side launch. d_in order: feat, rois, roi_batch_id, W1,b1, W2,b2, Wc,bc, Wr,br
// d_out: [cls 128x21 | reg 128x84] fp32, flat in return order.
// ---------------------------------------------------------------------------
extern "C" void kernel_launch(void* const* d_in, const int* in_sizes, int n_in,
                              void* d_out, int out_size, void* d_ws, size_t ws_size,
                              hipStream_t stream)
{
  const float* feat = (const float*)d_in[0];
  const float* rois = (const float*)d_in[1];
  const int*   rbid = (const int*)d_in[2];
  const float* W1   = (const float*)d_in[3];
  const float* b1   = (const float*)d_in[4];
  const float* W2   = (const float*)d_in[5];
  const float* b2   = (const float*)d_in[6];
  const float* Wc   = (const float*)d_in[7];
  const float* bc   = (const float*)d_in[8];
  const float* Wr   = (const float*)d_in[9];
  const float* br   = (const float*)d_in[10];
  float* out = (float*)d_out;

  constexpr int S1 = 8;                 // FC1 K-slices (2048 waves)
  constexpr int S2 = 4;                 // FC2 K-slices (1024 waves)

  // Workspace: X [128x25088] | h1 [128x4096] | h2 [128x4096] | partials [S1x128x4096]
  float* Xp   = (float*)d_ws;                       // 12.85 MB
  float* h1   = Xp   + (size_t)RROI * KFC1;         //  2 MB
  float* h2   = h1   + (size_t)RROI * NHID;         //  2 MB
  float* part = h2   + (size_t)RROI * NHID;         // 16 MB (reused by FC2)

  roi_pool_kernel<<<dim3(RROI, CCH), 64, 0, stream>>>(feat, rois, rbid, Xp);

  // FC1: X[128,25088] @ W1[25088,4096] (411 MB NT stream), split-K=8
  wmma_gemm128<KFC1, KFC1 / S1, NHID, false, false>
      <<<dim3(NHID / 64, S1), 128, 0, stream>>>(Xp, W1, b1, part);
  reduce_bias_relu<NHID, S1>
      <<<(RROI * NHID / 4 + 255) / 256, 256, 0, stream>>>(part, b1, h1);

  // FC2: h1 @ W2[4096,4096] (67 MB NT stream), split-K=4
  wmma_gemm128<NHID, NHID / S2, NHID, false, false>
      <<<dim3(NHID / 64, S2), 128, 0, stream>>>(h1, W2, b2, part);
  reduce_bias_relu<NHID, S2>
      <<<(RROI * NHID / 4 + 255) / 256, 256, 0, stream>>>(part, b2, h2);

  // Heads: single-slice, fused bias, masked edge tiles
  wmma_gemm128<NHID, NHID, 21, true, false>
      <<<dim3(1, 1), 128, 0, stream>>>(h2, Wc, bc, out);
  wmma_gemm128<NHID, NHID, 84, true, false>
      <<<dim3(2, 1), 128, 0, stream>>>(h2, Wr, br, out + (size_t)RROI * 21);
}